// HCSAGE_57294863729409
// MI455X (gfx1250) — compile-verified
//
#include <hip/hip_runtime.h>
#include <hip/hip_bf16.h>
#include <math.h>

// CDNA5 / gfx1250: wave32. WMMA f32 16x16x4 fragments:
//   A (16x4): lane = M row (lane%16), vgpr0/1 = K { (lane/16)*2, +1 }
//   B (4x16): lane = N col (lane%16), vgpr0/1 = K { (lane/16)*2, +1 }
//   C/D (16x16): lane = N (lane%16), vgpr r = M = r + 8*(lane/16)
typedef __attribute__((ext_vector_type(2))) float v2f;
typedef __attribute__((ext_vector_type(8))) float v8f;

#define D_IN  12
#define D_HID 64
#define D_OUTC 13

// ---------------- utility kernels ----------------

__global__ void hcsage_zero(float* __restrict__ p, long n) {
    long i = (long)blockIdx.x * blockDim.x + threadIdx.x;
    long stride = (long)gridDim.x * blockDim.x;
    for (; i < n; i += stride) p[i] = 0.0f;
}

__global__ void hcsage_degree(const int* __restrict__ dst, float* __restrict__ deg, int E) {
    int e = blockIdx.x * blockDim.x + threadIdx.x;
    if (e < E) atomicAdd(&deg[dst[e]], 1.0f);
}

__global__ void hcsage_invdeg(float* __restrict__ deg, int N) {
    int i = blockIdx.x * blockDim.x + threadIdx.x;
    if (i < N) deg[i] = 1.0f / fmaxf(deg[i], 1.0f);
}

// scatter-add x[src] rows (12 ch) into agg1[dst]
__global__ void hcsage_agg12(const int* __restrict__ src, const int* __restrict__ dst,
                             const float* __restrict__ x, float* __restrict__ agg, int E) {
    int e = blockIdx.x * blockDim.x + threadIdx.x;
    if (e >= E) return;
    const float* xr = x + (size_t)src[e] * D_IN;
    float* ar = agg + (size_t)dst[e] * D_IN;
#pragma unroll
    for (int c = 0; c < D_IN; ++c) atomicAdd(&ar[c], xr[c]);
}

// scatter-add h[src] rows (64 ch) into agg2[dst]; 16 threads/edge, float4 each
__global__ void hcsage_agg64(const int* __restrict__ src, const int* __restrict__ dst,
                             const float* __restrict__ h, float* __restrict__ agg, int E) {
    long t = (long)blockIdx.x * blockDim.x + threadIdx.x;
    if (t >= (long)E * 16) return;
    int e  = (int)(t >> 4);
    int c4 = (int)(t & 15);
    const float4 v = *(const float4*)(h + (size_t)src[e] * D_HID + c4 * 4);
    float* ar = agg + (size_t)dst[e] * D_HID + c4 * 4;
    atomicAdd(&ar[0], v.x);
    atomicAdd(&ar[1], v.y);
    atomicAdd(&ar[2], v.z);
    atomicAdd(&ar[3], v.w);
}

// ---------------- layer 1: h = relu([agg1*invd | x] @ [Wl1|Wr1]^T + b1) ----------------
// wave = one 16-node tile; combined K = 24 (12 agg + 12 x); 4 N-tiles of 16 -> D_HID=64
__global__ void __launch_bounds__(256)
hcsage_lin1(const float* __restrict__ agg1, const float* __restrict__ x,
            const float* __restrict__ invd,
            const float* __restrict__ Wl1, const float* __restrict__ Wr1,
            const float* __restrict__ b1, float* __restrict__ h, int N) {
    __shared__ float wc[24 * 64];           // wc[k][n]: combined weights, 6 KB
    for (int idx = threadIdx.x; idx < 24 * 64; idx += blockDim.x) {
        int k = idx >> 6, n = idx & 63;
        wc[idx] = (k < D_IN) ? Wl1[n * D_IN + k] : Wr1[n * D_IN + (k - D_IN)];
    }
    __syncthreads();

    int wave = threadIdx.x >> 5;
    int lane = threadIdx.x & 31;
    int base = (blockIdx.x * 8 + wave) * 16;
    if (base >= N) return;                  // wave-uniform: EXEC stays all-ones

    int m    = lane & 15;
    int half = lane >> 4;
    int kh   = half * 2;
    int nl   = lane & 15;
    int row  = base + m;
    float scale = invd[row];
    const float* ag = agg1 + (size_t)row * D_IN;
    const float* xr = x    + (size_t)row * D_IN;

    v2f a[6];
#pragma unroll
    for (int s = 0; s < 3; ++s) {           // K 0..11: mean-aggregated neighbors
        int k0 = s * 4 + kh;
        a[s].x = ag[k0] * scale;
        a[s].y = ag[k0 + 1] * scale;
    }
#pragma unroll
    for (int s = 3; s < 6; ++s) {           // K 12..23: root features
        int k0 = s * 4 + kh - D_IN;
        a[s].x = xr[k0];
        a[s].y = xr[k0 + 1];
    }

#pragma unroll
    for (int t = 0; t < 4; ++t) {
        v8f c = {};
#pragma unroll
        for (int s = 0; s < 6; ++s) {
            int k0 = s * 4 + kh;
            v2f b;
            b.x = wc[k0 * 64 + t * 16 + nl];
            b.y = wc[(k0 + 1) * 64 + t * 16 + nl];
            c = __builtin_amdgcn_wmma_f32_16x16x4_f32(
                    false, a[s], false, b, (short)0, c, false, false);
        }
        float bias = b1[t * 16 + nl];
#pragma unroll
        for (int r = 0; r < 8; ++r) {
            int mm = r + 8 * half;
            h[(size_t)(base + mm) * D_HID + t * 16 + nl] = fmaxf(c[r] + bias, 0.0f);
        }
    }
}

// ---------------- layer 2: oraw = sigmoid([agg2*invd | h] @ [Wl2|Wr2]^T + b2) ----------------
// combined K = 128; single N-tile (13 real cols padded to 16 with zero weights)
__global__ void __launch_bounds__(256)
hcsage_lin2(const float* __restrict__ agg2, const float* __restrict__ h,
            const float* __restrict__ invd,
            const float* __restrict__ Wl2, const float* __restrict__ Wr2,
            const float* __restrict__ b2, float* __restrict__ oraw, int N) {
    __shared__ float wc[128 * 16];          // wc[k][n], 8 KB
    for (int idx = threadIdx.x; idx < 128 * 16; idx += blockDim.x) {
        int k = idx >> 4, n = idx & 15;
        float v = 0.0f;
        if (n < D_OUTC) v = (k < D_HID) ? Wl2[n * D_HID + k] : Wr2[n * D_HID + (k - D_HID)];
        wc[idx] = v;
    }
    __syncthreads();

    int wave = threadIdx.x >> 5;
    int lane = threadIdx.x & 31;
    int base = (blockIdx.x * 8 + wave) * 16;
    if (base >= N) return;

    int m    = lane & 15;
    int half = lane >> 4;
    int kh   = half * 2;
    int nl   = lane & 15;
    int row  = base + m;
    float scale = invd[row];
    const float* ag = agg2 + (size_t)row * D_HID;
    const float* hr = h    + (size_t)row * D_HID;

    v8f c = {};
#pragma unroll
    for (int s = 0; s < 16; ++s) {          // K 0..63: mean-aggregated hidden
        int k0 = s * 4 + kh;
        v2f a; a.x = ag[k0] * scale; a.y = ag[k0 + 1] * scale;
        v2f b; b.x = wc[k0 * 16 + nl]; b.y = wc[(k0 + 1) * 16 + nl];
        c = __builtin_amdgcn_wmma_f32_16x16x4_f32(
                false, a, false, b, (short)0, c, false, false);
    }
#pragma unroll
    for (int s = 16; s < 32; ++s) {         // K 64..127: root hidden
        int k0 = s * 4 + kh;
        v2f a; a.x = hr[k0 - D_HID]; a.y = hr[k0 - D_HID + 1];
        v2f b; b.x = wc[k0 * 16 + nl]; b.y = wc[(k0 + 1) * 16 + nl];
        c = __builtin_amdgcn_wmma_f32_16x16x4_f32(
                false, a, false, b, (short)0, c, false, false);
    }

    float bias = (nl < D_OUTC) ? b2[nl] : 0.0f;
#pragma unroll
    for (int r = 0; r < 8; ++r) {
        int mm = r + 8 * half;
        float z = c[r] + bias;
        float o = (nl < D_OUTC) ? (1.0f / (1.0f + __expf(-z))) : 0.0f;
        oraw[(size_t)(base + mm) * 16 + nl] = o;
    }
}

// ---------------- hierarchy constraint: out[n,i] = max_j R[i,j]*o[n,j] ----------------
__global__ void hcsage_constr(const float* __restrict__ oraw, const float* __restrict__ R,
                              float* __restrict__ out, int N) {
    __shared__ float rs[D_OUTC * D_OUTC];
    for (int i = threadIdx.x; i < D_OUTC * D_OUTC; i += blockDim.x) rs[i] = R[i];
    __syncthreads();
    int n = blockIdx.x * blockDim.x + threadIdx.x;
    if (n >= N) return;
    float o[D_OUTC];
    const float* orow = oraw + (size_t)n * 16;
#pragma unroll
    for (int j = 0; j < D_OUTC; ++j) o[j] = orow[j];
#pragma unroll
    for (int i = 0; i < D_OUTC; ++i) {
        float mx = rs[i * D_OUTC] * o[0];
#pragma unroll
        for (int j = 1; j < D_OUTC; ++j) mx = fmaxf(mx, rs[i * D_OUTC + j] * o[j]);
        out[(size_t)n * D_OUTC + i] = mx;
    }
}

// ---------------- launcher ----------------

extern "C" void kernel_launch(void* const* d_in, const int* in_sizes, int n_in,
                              void* d_out, int out_size, void* d_ws, size_t ws_size,
                              hipStream_t stream) {
    const float* x   = (const float*)d_in[0];
    const int*   ei  = (const int*)  d_in[1];
    const float* Wl1 = (const float*)d_in[2];
    const float* Wr1 = (const float*)d_in[3];
    const float* b1  = (const float*)d_in[4];
    const float* Wl2 = (const float*)d_in[5];
    const float* Wr2 = (const float*)d_in[6];
    const float* b2  = (const float*)d_in[7];
    const float* R   = (const float*)d_in[8];

    const int N = in_sizes[0] / D_IN;       // 100,000
    const int E = in_sizes[1] / 2;          // 3,200,000
    const int* src = ei;
    const int* dst = ei + E;

    // workspace layout (floats): [deg N][agg1 N*12][agg2 N*64][h N*64][oraw N*16]
    float* ws   = (float*)d_ws;
    float* deg  = ws;
    float* agg1 = deg  + (size_t)N;
    float* agg2 = agg1 + (size_t)N * D_IN;
    float* h    = agg2 + (size_t)N * D_HID;
    float* oraw = h    + (size_t)N * D_HID;

    const int TB = 256;
    // 1) zero deg + agg1 + agg2 (contiguous N*77 floats)
    hcsage_zero<<<2048, TB, 0, stream>>>(deg, (long)N * (1 + D_IN + D_HID));
    // 2) degree + reciprocal
    hcsage_degree<<<(E + TB - 1) / TB, TB, 0, stream>>>(dst, deg, E);
    hcsage_invdeg<<<(N + TB - 1) / TB, TB, 0, stream>>>(deg, N);
    // 3) layer-1 aggregation (scatter)
    hcsage_agg12<<<(E + TB - 1) / TB, TB, 0, stream>>>(src, dst, x, agg1, E);
    // 4) layer-1 dense (WMMA f32) + ReLU
    int tiles = (N + 15) / 16;
    int lblocks = (tiles + 7) / 8;
    hcsage_lin1<<<lblocks, TB, 0, stream>>>(agg1, x, deg, Wl1, Wr1, b1, h, N);
    // 5) layer-2 aggregation (scatter, float4)
    long t64 = (long)E * 16;
    hcsage_agg64<<<(int)((t64 + TB - 1) / TB), TB, 0, stream>>>(src, dst, h, agg2, E);
    // 6) layer-2 dense (WMMA f32) + sigmoid
    hcsage_lin2<<<lblocks, TB, 0, stream>>>(agg2, h, deg, Wl2, Wr2, b2, oraw, N);
    // 7) hierarchy constraint
    hcsage_constr<<<(N + TB - 1) / TB, TB, 0, stream>>>(oraw, R, (float*)d_out, N);
}